// LRU_88270167867994
// MI455X (gfx1250) — compile-verified
//
#include <hip/hip_runtime.h>

// ---------------------------------------------------------------------------
// LRU forward for MI455X (gfx1250, wave32, WMMA).
//   B=8, T=4096, D_IN=D_OUT=256, N=512
// Pipeline:
//   P1: lam[n] = exp(-exp(nu))*(cos(exp(th)), sin(exp(th)))
//   P2: Wg[2N x D_IN]  = gamma[n] * {B_real ; B_imag}
//   P3: Wout[D_OUT x 1280] = {2*C_real ; -2*C_imag ; D}
//   A : BuX[32768 x 1024] = u @ Wg^T                 (f32 WMMA 16x16x4)
//   B : local chunk scans (chunk=256) in place over BuX, emit chunk-end carries
//   C : carry scan across 16 chunks (lam^256 via 8 squarings)
//   D : apply carries: x_t += lam^(t+1) * carry
//   E : y = X @ Wout[:, :1024]^T + u @ Wout[:, 1024:]^T   (f32 WMMA 16x16x4)
// ---------------------------------------------------------------------------

typedef float v2f __attribute__((ext_vector_type(2)));
typedef float v8f __attribute__((ext_vector_type(8)));

#define T_LEN   4096
#define BATCH   8
#define D_IN    256
#define D_OUT   256
#define N_ST    512
#define ROWS    (BATCH * T_LEN)      // 32768
#define KW      (2 * N_ST)           // 1024 (real plane | imag plane)
#define CH_LEN  256
#define N_CH    (T_LEN / CH_LEN)     // 16
#define KOUT    1280                 // 512 + 512 + 256

// ---------------------------- prep kernels ---------------------------------

__global__ void lru_prep_lam(const float* __restrict__ nu_log,
                             const float* __restrict__ theta_log,
                             float* __restrict__ lam) {
    int n = blockIdx.x * blockDim.x + threadIdx.x;
    if (n < N_ST) {
        float lmod = expf(-expf(nu_log[n]));
        float th   = expf(theta_log[n]);
        lam[2 * n + 0] = lmod * cosf(th);
        lam[2 * n + 1] = lmod * sinf(th);
    }
}

__global__ void lru_prep_wg(const float* __restrict__ Br,
                            const float* __restrict__ Bi,
                            const float* __restrict__ gamma_log,
                            float* __restrict__ Wg) {
    int i = blockIdx.x * blockDim.x + threadIdx.x;     // over KW * D_IN
    if (i < KW * D_IN) {
        int n = i / D_IN;
        int k = i - n * D_IN;
        int nn = (n < N_ST) ? n : (n - N_ST);
        float g = expf(gamma_log[nn]);
        float v = (n < N_ST) ? Br[nn * D_IN + k] : Bi[nn * D_IN + k];
        Wg[i] = g * v;
    }
}

__global__ void lru_prep_wout(const float* __restrict__ Cr,
                              const float* __restrict__ Ci,
                              const float* __restrict__ Dm,
                              float* __restrict__ Wout) {
    int i = blockIdx.x * blockDim.x + threadIdx.x;     // over D_OUT * KOUT
    if (i < D_OUT * KOUT) {
        int o = i / KOUT;
        int k = i - o * KOUT;
        float v;
        if (k < 512)       v =  2.0f * Cr[o * N_ST + k];
        else if (k < 1024) v = -2.0f * Ci[o * N_ST + (k - 512)];
        else               v =  Dm[o * D_IN + (k - 1024)];
        Wout[i] = v;
    }
}

// ------------------------- WMMA 16x64 tile helper --------------------------
// A: row-major [M x lda] activations, W: row-major [Ncols x ldw] weights.
// Computes acc[j] += A[m0:m0+16, 0:K] * W[n0+16j : n0+16j+16, 0:K]^T
// using V_WMMA_F32_16X16X4_F32. Per the ISA 16x4 f32 A layout:
//   lane<16:  a = {A[m0+lane][k], A[m0+lane][k+1]}
//   lane>=16: a = {A[m0+lane-16][k+2], A[m0+lane-16][k+3]}
// (symmetric for B) -> one contiguous float2 load per operand per step.

__device__ __forceinline__ void wmma_tile_16x64(const float* __restrict__ A, int lda,
                                                const float* __restrict__ W, int ldw,
                                                int K, int m0, int n0, v8f acc[4]) {
    int lane = threadIdx.x & 31;
    int half = lane >> 4;
    int lh   = lane & 15;
    const float* arow = A + (size_t)(m0 + lh) * lda + 2 * half;
    const float* wrow = W + (size_t)(n0 + lh) * ldw + 2 * half;
    #pragma unroll 4
    for (int k = 0; k < K; k += 4) {
        v2f av = *(const v2f*)(arow + k);
        #pragma unroll
        for (int j = 0; j < 4; ++j) {
            v2f bv = *(const v2f*)(wrow + (size_t)(16 * j) * ldw + k);
            acc[j] = __builtin_amdgcn_wmma_f32_16x16x4_f32(
                false, av, false, bv, (short)0, acc[j], false, false);
        }
    }
}

__device__ __forceinline__ void store_tile_16x64(float* __restrict__ Out, int ldo,
                                                 int m0, int n0, const v8f acc[4]) {
    int lane = threadIdx.x & 31;
    int half = lane >> 4;
    int lh   = lane & 15;
    #pragma unroll
    for (int j = 0; j < 4; ++j) {
        int col = n0 + 16 * j + lh;
        #pragma unroll
        for (int v = 0; v < 8; ++v) {
            int row = m0 + v + 8 * half;
            Out[(size_t)row * ldo + col] = acc[j][v];
        }
    }
}

// -------------------------- GEMM kernels (WMMA) ----------------------------

// BuX[32768 x 1024] = u[32768 x 256] @ Wg[1024 x 256]^T
// 8 waves/block, each wave one 16x64 tile. waves = 2048 (M) * 16 (Ngrp) = 32768.
__global__ void lru_gemm_bu(const float* __restrict__ u,
                            const float* __restrict__ Wg,
                            float* __restrict__ BuX) {
    int wave = blockIdx.x * (blockDim.x >> 5) + (threadIdx.x >> 5);
    int mt = wave >> 4;
    int ng = wave & 15;
    int m0 = mt * 16, n0 = ng * 64;
    v8f acc[4] = {};
    wmma_tile_16x64(u, D_IN, Wg, D_IN, D_IN, m0, n0, acc);
    store_tile_16x64(BuX, KW, m0, n0, acc);
}

// y[32768 x 256] = X[32768 x 1024] @ Wout[:, :1024]^T + u @ Wout[:, 1024:]^T
// waves = 2048 (M) * 4 (Ngrp of 64) = 8192.
__global__ void lru_gemm_out(const float* __restrict__ X,
                             const float* __restrict__ u,
                             const float* __restrict__ Wout,
                             float* __restrict__ y) {
    int wave = blockIdx.x * (blockDim.x >> 5) + (threadIdx.x >> 5);
    int mt = wave >> 2;
    int ng = wave & 3;
    int m0 = mt * 16, n0 = ng * 64;
    v8f acc[4] = {};
    wmma_tile_16x64(X, KW, Wout, KOUT, KW, m0, n0, acc);
    wmma_tile_16x64(u, D_IN, Wout + 1024, KOUT, D_IN, m0, n0, acc);
    store_tile_16x64(y, D_OUT, m0, n0, acc);
}

// ----------------------------- scan kernels --------------------------------

// Local in-chunk scans, in place over BuX. tid -> (b, ch, n). 65536 threads.
__global__ void lru_scan_local(float* __restrict__ BuX,
                               const float* __restrict__ lam,
                               float* __restrict__ carry) {
    int tid = blockIdx.x * blockDim.x + threadIdx.x;
    int n  = tid & (N_ST - 1);
    int ch = (tid >> 9) & (N_CH - 1);
    int b  = tid >> 13;
    float lr = lam[2 * n], li = lam[2 * n + 1];
    size_t base = ((size_t)b * T_LEN + (size_t)ch * CH_LEN) * KW + n;
    float xr = 0.f, xi = 0.f;
    for (int t = 0; t < CH_LEN; ++t) {
        float br = BuX[base], bi = BuX[base + N_ST];
        float nr = fmaf(lr, xr, fmaf(-li, xi, br));
        float ni = fmaf(lr, xi, fmaf( li, xr, bi));
        xr = nr; xi = ni;
        BuX[base] = xr; BuX[base + N_ST] = xi;
        base += KW;
    }
    size_t cidx = 2 * (((size_t)b * N_CH + ch) * N_ST + n);
    carry[cidx] = xr; carry[cidx + 1] = xi;
}

// Scan chunk-end values into incoming prefixes P_c = S_{c-1}. 4096 threads.
__global__ void lru_scan_carry(float* __restrict__ carry,
                               const float* __restrict__ lam) {
    int tid = blockIdx.x * blockDim.x + threadIdx.x;
    int n = tid & (N_ST - 1);
    int b = tid >> 9;
    float lr = lam[2 * n], li = lam[2 * n + 1];
    // lam^256 via 8 complex squarings
    float pr = lr, pi = li;
    #pragma unroll
    for (int s = 0; s < 8; ++s) {
        float nr = pr * pr - pi * pi;
        float ni = 2.f * pr * pi;
        pr = nr; pi = ni;
    }
    float sr = 0.f, si = 0.f;
    for (int ch = 0; ch < N_CH; ++ch) {
        size_t cidx = 2 * (((size_t)b * N_CH + ch) * N_ST + n);
        float er = carry[cidx], ei = carry[cidx + 1];
        carry[cidx] = sr; carry[cidx + 1] = si;            // incoming prefix
        float nr = fmaf(pr, sr, fmaf(-pi, si, er));        // S_c = e_c + lam^L * S
        float ni = fmaf(pr, si, fmaf( pi, sr, ei));
        sr = nr; si = ni;
    }
}

// x_t(global) = x_t(local) + lam^(t+1) * P_c. 65536 threads (chunk 0 is a no-op).
__global__ void lru_scan_apply(float* __restrict__ X,
                               const float* __restrict__ lam,
                               const float* __restrict__ carry) {
    int tid = blockIdx.x * blockDim.x + threadIdx.x;
    int n  = tid & (N_ST - 1);
    int ch = (tid >> 9) & (N_CH - 1);
    int b  = tid >> 13;
    if (ch == 0) return;                                   // prefix is exactly zero
    size_t cidx = 2 * (((size_t)b * N_CH + ch) * N_ST + n);
    float cr = carry[cidx], ci = carry[cidx + 1];
    float lr = lam[2 * n], li = lam[2 * n + 1];
    size_t base = ((size_t)b * T_LEN + (size_t)ch * CH_LEN) * KW + n;
    for (int t = 0; t < CH_LEN; ++t) {
        float nr = cr * lr - ci * li;
        float ni = cr * li + ci * lr;
        cr = nr; ci = ni;
        X[base]        += cr;
        X[base + N_ST] += ci;
        base += KW;
    }
}

// ------------------------------ launcher -----------------------------------

extern "C" void kernel_launch(void* const* d_in, const int* in_sizes, int n_in,
                              void* d_out, int out_size, void* d_ws, size_t ws_size,
                              hipStream_t stream) {
    (void)in_sizes; (void)n_in; (void)out_size; (void)ws_size;

    const float* u_in      = (const float*)d_in[0];
    const float* nu_log    = (const float*)d_in[1];
    const float* theta_log = (const float*)d_in[2];
    const float* gamma_log = (const float*)d_in[3];
    const float* B_real    = (const float*)d_in[4];
    const float* B_imag    = (const float*)d_in[5];
    const float* C_real    = (const float*)d_in[6];
    const float* C_imag    = (const float*)d_in[7];
    const float* Dm        = (const float*)d_in[8];
    float* y = (float*)d_out;

    float* ws = (float*)d_ws;
    size_t off = 0;
    float* Wg    = ws + off; off += (size_t)KW * D_IN;        // 1 MB
    float* Wout  = ws + off; off += (size_t)D_OUT * KOUT;     // 1.25 MB
    float* lam   = ws + off; off += 2 * N_ST;
    off = (off + 1023) & ~(size_t)1023;
    float* BuX   = ws + off; off += (size_t)ROWS * KW;        // 128 MB (state, in place)
    float* carry = ws + off; off += (size_t)BATCH * N_CH * N_ST * 2;

    lru_prep_lam <<<2, 256, 0, stream>>>(nu_log, theta_log, lam);
    lru_prep_wg  <<<(KW * D_IN) / 256, 256, 0, stream>>>(B_real, B_imag, gamma_log, Wg);
    lru_prep_wout<<<(D_OUT * KOUT) / 256, 256, 0, stream>>>(C_real, C_imag, Dm, Wout);

    lru_gemm_bu  <<<4096, 256, 0, stream>>>(u_in, Wg, BuX);       // 32768 waves
    lru_scan_local<<<256, 256, 0, stream>>>(BuX, lam, carry);
    lru_scan_carry<<<16, 256, 0, stream>>>(carry, lam);
    lru_scan_apply<<<256, 256, 0, stream>>>(BuX, lam, carry);
    lru_gemm_out <<<1024, 256, 0, stream>>>(BuX, u_in, Wout, y);  // 8192 waves
}